// Net_90331752170289
// MI455X (gfx1250) — compile-verified
//
#include <hip/hip_runtime.h>
#include <hip/hip_bf16.h>
#include <math.h>

#define B_  64
#define L_  2048
#define DM  12
#define DI  24
#define DS_ 16
#define DC  4
#define NT  (B_ * L_)   // 131072 tokens

typedef float v2f __attribute__((ext_vector_type(2)));
typedef float v8f __attribute__((ext_vector_type(8)));

#if defined(__AMDGCN__) && __has_builtin(__builtin_amdgcn_wmma_f32_16x16x4_f32)
#define HAS_WMMA_F32 1
#else
#define HAS_WMMA_F32 0
#endif

__device__ __forceinline__ float silu_f(float v) { return v / (1.f + expf(-v)); }
__device__ __forceinline__ float softplus_f(float v) { return (v > 20.f) ? v : log1pf(expf(v)); }

// ---------------------------------------------------------------------------
// K1: in_proj -> causal depthwise conv(DC=4) -> SiLU -> x_proj -> delta
// one thread per token; stores xi(24), delta(24), Bm(16), Cm(16)
// ---------------------------------------------------------------------------
__global__ void pre_kernel(const float* __restrict__ x, const float* __restrict__ Win,
                           const float* __restrict__ convw, const float* __restrict__ convb,
                           const float* __restrict__ Wxp, const float* __restrict__ dtw,
                           const float* __restrict__ dtb,
                           float* __restrict__ xi_o, float* __restrict__ delta_o,
                           float* __restrict__ Bm_o, float* __restrict__ Cm_o) {
  int t = blockIdx.x * blockDim.x + threadIdx.x;
  if (t >= NT) return;
  int l = t % L_;
  float xr[DC][DM];
  for (int k = 0; k < DC; ++k) {
    int ll = l - (DC - 1) + k;
    for (int j = 0; j < DM; ++j)
      xr[k][j] = (ll >= 0) ? x[(size_t)(t - (DC - 1) + k) * DM + j] : 0.f;
  }
  float xi[DI];
  for (int d = 0; d < DI; ++d) {
    float c = convb[d];
    for (int k = 0; k < DC; ++k) {
      float r = 0.f;
      for (int j = 0; j < DM; ++j) r += xr[k][j] * Win[d * DM + j];
      c += r * convw[d * DC + k];
    }
    xi[d] = silu_f(c);
  }
  // x_proj: row 0 = dt, rows 1..16 = B, rows 17..32 = C
  float dt0 = 0.f;
  for (int d = 0; d < DI; ++d) dt0 += xi[d] * Wxp[d];
  for (int n = 0; n < DS_; ++n) {
    float sb = 0.f, sc = 0.f;
    for (int d = 0; d < DI; ++d) {
      sb += xi[d] * Wxp[(1 + n) * DI + d];
      sc += xi[d] * Wxp[(1 + DS_ + n) * DI + d];
    }
    Bm_o[(size_t)t * DS_ + n] = sb;
    Cm_o[(size_t)t * DS_ + n] = sc;
  }
  for (int d = 0; d < DI; ++d) {
    xi_o[(size_t)t * DI + d] = xi[d];
    delta_o[(size_t)t * DI + d] = softplus_f(dt0 * dtw[d] + dtb[d]);
  }
}

// ---------------------------------------------------------------------------
// K2: selective scan. one wave32 per (b,d); lanes 0..15 hold states n.
// h = h*exp(delta*A) + delta*B*xi ; y = <h,C> + xi*Dp  (shuffle reduction)
// ---------------------------------------------------------------------------
__global__ void scan_kernel(const float* __restrict__ xi, const float* __restrict__ delta,
                            const float* __restrict__ Bm, const float* __restrict__ Cm,
                            const float* __restrict__ A_log, const float* __restrict__ Dp,
                            float* __restrict__ y) {
  int bd = blockIdx.x;                 // b*DI + d
  int b = bd / DI, d = bd % DI;
  int lane = threadIdx.x;              // 0..31 (halves do duplicate work)
  int n = lane & (DS_ - 1);
  float A = -expf(A_log[d * DS_ + n]);
  float Dpd = Dp[d];
  float h = 0.f;
  size_t base = (size_t)b * L_;
  for (int l = 0; l < L_; ++l) {
    size_t tk = base + l;
    if (l + 16 < L_) {
      __builtin_prefetch(&Bm[(tk + 16) * DS_ + n], 0, 1);
      __builtin_prefetch(&delta[(tk + 16) * DI + d], 0, 1);
    }
    float ds = delta[tk * DI + d];
    float xs = xi[tk * DI + d];
    float Bn = Bm[tk * DS_ + n];
    float Cn = Cm[tk * DS_ + n];
    h = h * expf(ds * A) + ds * Bn * xs;
    float p = h * Cn;
    p += __shfl_xor(p, 8, 16);
    p += __shfl_xor(p, 4, 16);
    p += __shfl_xor(p, 2, 16);
    p += __shfl_xor(p, 1, 16);
    if (lane == 0) y[tk * DI + d] = p + xs * Dpd;
  }
}

// ---------------------------------------------------------------------------
// K3a: gating (recompute z), out_proj, LayerNorm -> h
// ---------------------------------------------------------------------------
__global__ void gate_kernel(const float* __restrict__ x, const float* __restrict__ Win,
                            const float* __restrict__ yb, const float* __restrict__ Wout,
                            const float* __restrict__ g, const float* __restrict__ be,
                            float* __restrict__ hbuf) {
  int t = blockIdx.x * blockDim.x + threadIdx.x;
  if (t >= NT) return;
  const float* xr = x + (size_t)t * DM;
  float yg[DI];
  for (int e = 0; e < DI; ++e) {
    float zv = 0.f;
    for (int j = 0; j < DM; ++j) zv += xr[j] * Win[(DI + e) * DM + j];
    yg[e] = yb[(size_t)t * DI + e] * silu_f(zv);
  }
  float m[DM];
  float mean = 0.f;
  for (int d = 0; d < DM; ++d) {
    float s = 0.f;
    for (int e = 0; e < DI; ++e) s += yg[e] * Wout[d * DI + e];
    m[d] = s;
    mean += s;
  }
  mean *= (1.f / DM);
  float var = 0.f;
  for (int d = 0; d < DM; ++d) { float dv = m[d] - mean; var += dv * dv; }
  var *= (1.f / DM);
  float inv = rsqrtf(var + 1e-12f);
  for (int d = 0; d < DM; ++d)
    hbuf[(size_t)t * DM + d] = (m[d] - mean) * inv * g[d] + be[d];
}

// ---------------------------------------------------------------------------
// K3b: FFN with V_WMMA_F32_16X16X4_F32. One wave per 16 tokens.
// GEMM1: (16 x 48) = h(16x12) @ W1^T  -> GELU -> LDS transpose
// GEMM2: (16 x 16) = ff(16x48) @ W2^T(48x12 pad16) -> +h -> LN -> x_mamba
// ---------------------------------------------------------------------------
__global__ void __launch_bounds__(256) ffn_kernel(const float* __restrict__ hbuf,
                                                  const float* __restrict__ w1, const float* __restrict__ b1,
                                                  const float* __restrict__ w2, const float* __restrict__ b2,
                                                  const float* __restrict__ g, const float* __restrict__ be,
                                                  float* __restrict__ xm) {
  __shared__ float lds[8][16][49];
  int lane = threadIdx.x & 31;
  int wib = threadIdx.x >> 5;
  size_t t0 = ((size_t)blockIdx.x * 8 + wib) * 16;
  int m = lane & 15;                  // A: row (token); B/C/D: column
  int koff = (lane >> 4) << 1;        // K sub-offset: lanes16-31 hold K+2,K+3

  // A-operand chunks (16x4 each) for K = 0..11
  v2f a[3];
  for (int kc = 0; kc < 3; ++kc) {
    const float* p = hbuf + (t0 + m) * DM + kc * 4 + koff;
    a[kc] = (v2f){p[0], p[1]};
  }

  for (int nt = 0; nt < 3; ++nt) {
    int n = nt * 16 + m;              // output feature in [0,48)
    float bias = b1[n];
    v8f acc;
    for (int v = 0; v < 8; ++v) acc[v] = bias;
#if HAS_WMMA_F32
    for (int kc = 0; kc < 3; ++kc) {
      const float* q = w1 + n * DM + kc * 4 + koff;
      v2f bm = (v2f){q[0], q[1]};
      acc = __builtin_amdgcn_wmma_f32_16x16x4_f32(false, a[kc], false, bm,
                                                  (short)0, acc, false, false);
    }
#else
    for (int v = 0; v < 8; ++v) {
      int tok = v + ((lane >> 4) << 3);
      float s = bias;
      for (int k = 0; k < DM; ++k) s += hbuf[(t0 + tok) * DM + k] * w1[n * DM + k];
      acc[v] = s;
    }
#endif
    // exact GELU, store tile (D layout) into LDS for the GEMM2 A-operand
    for (int v = 0; v < 8; ++v) {
      int tok = v + ((lane >> 4) << 3);
      float f = acc[v];
      lds[wib][tok][n] = 0.5f * f * (1.f + erff(f * 0.70710678118654752f));
    }
  }
  __syncthreads();

  int n2 = m;                          // output feature (valid < 12)
  float bias2 = (n2 < DM) ? b2[n2] : 0.f;
  v8f acc2;
  for (int v = 0; v < 8; ++v) acc2[v] = bias2;
#if HAS_WMMA_F32
  for (int kc = 0; kc < 12; ++kc) {
    int kb = kc * 4 + koff;
    v2f a2 = (v2f){lds[wib][m][kb], lds[wib][m][kb + 1]};
    v2f bv;
    if (n2 < DM) {
      const float* q = w2 + n2 * 48 + kb;
      bv = (v2f){q[0], q[1]};
    } else {
      bv = (v2f){0.f, 0.f};
    }
    acc2 = __builtin_amdgcn_wmma_f32_16x16x4_f32(false, a2, false, bv,
                                                 (short)0, acc2, false, false);
  }
#else
  for (int v = 0; v < 8; ++v) {
    int tok = v + ((lane >> 4) << 3);
    float s = bias2;
    if (n2 < DM)
      for (int k = 0; k < 48; ++k) s += lds[wib][tok][k] * w2[n2 * 48 + k];
    acc2[v] = (n2 < DM) ? s : 0.f;
  }
#endif

  // residual + LayerNorm over the 12 features, in D layout (width-16 shuffles)
  for (int v = 0; v < 8; ++v) {
    int tok = v + ((lane >> 4) << 3);
    float val = acc2[v];
    if (n2 < DM) val += hbuf[(t0 + tok) * DM + n2]; else val = 0.f;
    float s1 = val;
    s1 += __shfl_xor(s1, 8, 16);
    s1 += __shfl_xor(s1, 4, 16);
    s1 += __shfl_xor(s1, 2, 16);
    s1 += __shfl_xor(s1, 1, 16);
    float mean = s1 * (1.f / DM);
    float dv = (n2 < DM) ? (val - mean) : 0.f;
    float s2 = dv * dv;
    s2 += __shfl_xor(s2, 8, 16);
    s2 += __shfl_xor(s2, 4, 16);
    s2 += __shfl_xor(s2, 2, 16);
    s2 += __shfl_xor(s2, 1, 16);
    float inv = rsqrtf(s2 * (1.f / DM) + 1e-12f);
    if (n2 < DM)
      xm[(t0 + tok) * DM + n2] = (val - mean) * inv * g[n2] + be[n2];
  }
}

// ---------------------------------------------------------------------------
// K4: |FFT_2048| over L for each (b, channel). Stockham autosort in LDS.
// ---------------------------------------------------------------------------
__global__ void __launch_bounds__(256) fft_kernel(const float* __restrict__ x,
                                                  float* __restrict__ xf) {
  __shared__ float2 buf0[L_];
  __shared__ float2 buf1[L_];
  int bc = blockIdx.x;                // b*DM + ch
  int b = bc / DM, ch = bc % DM;
  for (int l = threadIdx.x; l < L_; l += 256)
    buf0[l] = make_float2(x[((size_t)b * L_ + l) * DM + ch], 0.f);
  __syncthreads();
  float2* src = buf0;
  float2* dst = buf1;
  int ncur = L_, s = 1;
  while (ncur > 1) {
    int mh = ncur >> 1;
    for (int i = threadIdx.x; i < (L_ / 2); i += 256) {
      int p = i / s;
      int q = i - p * s;
      float2 ua = src[q + s * p];
      float2 ub = src[q + s * (p + mh)];
      float ang = -6.283185307179586f * (float)p / (float)ncur;
      float wr = cosf(ang), wi = sinf(ang);
      float dr = ua.x - ub.x, di = ua.y - ub.y;
      dst[q + s * (2 * p)] = make_float2(ua.x + ub.x, ua.y + ub.y);
      dst[q + s * (2 * p + 1)] = make_float2(dr * wr - di * wi, dr * wi + di * wr);
    }
    __syncthreads();
    float2* t = src; src = dst; dst = t;
    ncur = mh; s <<= 1;
  }
  for (int k = threadIdx.x; k < L_; k += 256) {
    float2 v = src[k];
    xf[((size_t)b * L_ + k) * DM + ch] = sqrtf(v.x * v.x + v.y * v.y);
  }
}

// ---------------------------------------------------------------------------
// K5: conv1d 12->12, kernel 3, pad (1,1)
// ---------------------------------------------------------------------------
__global__ void cnn_kernel(const float* __restrict__ xf, const float* __restrict__ w,
                           const float* __restrict__ bias, float* __restrict__ out) {
  int t = blockIdx.x * blockDim.x + threadIdx.x;
  if (t >= NT) return;
  int l = t % L_;
  float in0[DM], in1[DM], in2[DM];
  for (int j = 0; j < DM; ++j) {
    in1[j] = xf[(size_t)t * DM + j];
    in0[j] = (l > 0) ? xf[(size_t)(t - 1) * DM + j] : 0.f;
    in2[j] = (l < L_ - 1) ? xf[(size_t)(t + 1) * DM + j] : 0.f;
  }
  for (int o = 0; o < DM; ++o) {
    float s = bias[o];
    for (int i = 0; i < DM; ++i) {
      const float* wp = w + (o * DM + i) * 3;
      s += in0[i] * wp[0] + in1[i] * wp[1] + in2[i] * wp[2];
    }
    out[(size_t)t * DM + o] = s;
  }
}

// ---------------------------------------------------------------------------
// K6: head MLP (24->12 elu ->20 ->1), mean over L, sigmoid. One block per b.
// ---------------------------------------------------------------------------
__global__ void __launch_bounds__(256) head_kernel(const float* __restrict__ xm,
                                                   const float* __restrict__ xc,
                                                   const float* __restrict__ w1, const float* __restrict__ bb1,
                                                   const float* __restrict__ w2, const float* __restrict__ bb2,
                                                   const float* __restrict__ w3, const float* __restrict__ bb3,
                                                   float* __restrict__ out) {
  __shared__ float red[256];
  int b = blockIdx.x;
  float accum = 0.f;
  for (int l = threadIdx.x; l < L_; l += 256) {
    size_t t = (size_t)b * L_ + l;
    float comb[2 * DM];
    for (int j = 0; j < DM; ++j) {
      comb[j] = xm[t * DM + j];
      comb[DM + j] = xc[t * DM + j];
    }
    float o1[DM];
    for (int j = 0; j < DM; ++j) {
      float s = bb1[j];
      for (int e = 0; e < 2 * DM; ++e) s += comb[e] * w1[j * (2 * DM) + e];
      o1[j] = (s > 0.f) ? s : (expf(s) - 1.f);   // elu
    }
    float o3 = bb3[0];
    for (int k = 0; k < 20; ++k) {
      float s = bb2[k];
      for (int j = 0; j < DM; ++j) s += o1[j] * w2[k * DM + j];
      o3 += s * w3[k];
    }
    accum += o3;
  }
  red[threadIdx.x] = accum;
  __syncthreads();
  for (int st = 128; st > 0; st >>= 1) {
    if (threadIdx.x < st) red[threadIdx.x] += red[threadIdx.x + st];
    __syncthreads();
  }
  if (threadIdx.x == 0) out[b] = 1.f / (1.f + expf(-red[0] / (float)L_));
}

// ---------------------------------------------------------------------------
extern "C" void kernel_launch(void* const* d_in, const int* in_sizes, int n_in,
                              void* d_out, int out_size, void* d_ws, size_t ws_size,
                              hipStream_t stream) {
  const float* x      = (const float*)d_in[0];
  const float* Win    = (const float*)d_in[1];
  const float* convw  = (const float*)d_in[2];
  const float* convb  = (const float*)d_in[3];
  const float* Wxp    = (const float*)d_in[4];
  const float* dtw    = (const float*)d_in[5];
  const float* dtb    = (const float*)d_in[6];
  const float* A_log  = (const float*)d_in[7];
  const float* Dp     = (const float*)d_in[8];
  const float* Wout   = (const float*)d_in[9];
  const float* ln1g   = (const float*)d_in[10];
  const float* ln1b   = (const float*)d_in[11];
  const float* fw1    = (const float*)d_in[12];
  const float* fb1    = (const float*)d_in[13];
  const float* fw2    = (const float*)d_in[14];
  const float* fb2    = (const float*)d_in[15];
  const float* flng   = (const float*)d_in[16];
  const float* flnb   = (const float*)d_in[17];
  const float* cnnw   = (const float*)d_in[18];
  const float* cnnb   = (const float*)d_in[19];
  const float* l1w    = (const float*)d_in[20];
  const float* l1b    = (const float*)d_in[21];
  const float* l2w    = (const float*)d_in[22];
  const float* l2b    = (const float*)d_in[23];
  const float* l3w    = (const float*)d_in[24];
  const float* l3b    = (const float*)d_in[25];
  float* out = (float*)d_out;

  float* ws = (float*)d_ws;
  float* xi    = ws;                        // NT*24
  float* delta = xi    + (size_t)NT * DI;   // NT*24
  float* Bm    = delta + (size_t)NT * DI;   // NT*16
  float* Cm    = Bm    + (size_t)NT * DS_;  // NT*16
  float* yb    = Cm    + (size_t)NT * DS_;  // NT*24
  float* hbuf  = yb    + (size_t)NT * DI;   // NT*12
  float* xf    = hbuf  + (size_t)NT * DM;   // NT*12
  float* xcnn  = xf    + (size_t)NT * DM;   // NT*12

  pre_kernel<<<(NT + 255) / 256, 256, 0, stream>>>(x, Win, convw, convb, Wxp, dtw, dtb,
                                                   xi, delta, Bm, Cm);
  scan_kernel<<<B_ * DI, 32, 0, stream>>>(xi, delta, Bm, Cm, A_log, Dp, yb);
  gate_kernel<<<(NT + 255) / 256, 256, 0, stream>>>(x, Win, yb, Wout, ln1g, ln1b, hbuf);
  ffn_kernel<<<(NT / 16) / 8, 256, 0, stream>>>(hbuf, fw1, fb1, fw2, fb2, flng, flnb, yb /*reuse as x_mamba*/);
  fft_kernel<<<B_ * DM, 256, 0, stream>>>(x, xf);
  cnn_kernel<<<(NT + 255) / 256, 256, 0, stream>>>(xf, cnnw, cnnb, xcnn);
  head_kernel<<<B_, 256, 0, stream>>>(yb, xcnn, l1w, l1b, l2w, l2b, l3w, l3b, out);
}